// GNNModel_67216238182901
// MI455X (gfx1250) — compile-verified
//
#include <hip/hip_runtime.h>

#define N_NODES 50000
#define N_EDGES 800000
#define HDIM    128
#define NB      64

typedef __attribute__((ext_vector_type(2))) float v2f;
typedef __attribute__((ext_vector_type(8))) float v8f;

// ---------------------------------------------------------------------------
// Degree / norm precompute
// ---------------------------------------------------------------------------
__global__ __launch_bounds__(256) void gcn_init_deg(float* __restrict__ deg) {
    int i = blockIdx.x * 256 + threadIdx.x;
    if (i < N_NODES) deg[i] = 1.0f;            // self-loop
}

__global__ __launch_bounds__(256) void gcn_count_deg(const int* __restrict__ dst,
                                                     float* __restrict__ deg) {
    int e = blockIdx.x * 256 + threadIdx.x;
    if (e < N_EDGES) atomicAdd(&deg[dst[e]], 1.0f);
}

__global__ __launch_bounds__(256) void gcn_dinv(const float* __restrict__ deg,
                                                float* __restrict__ dinv) {
    int i = blockIdx.x * 256 + threadIdx.x;
    if (i < N_NODES) dinv[i] = rsqrtf(fmaxf(deg[i], 1.0f));
}

__global__ __launch_bounds__(256) void gcn_edge_norm(const int* __restrict__ src,
                                                     const int* __restrict__ dst,
                                                     const float* __restrict__ dinv,
                                                     float* __restrict__ nrm) {
    int e = blockIdx.x * 256 + threadIdx.x;
    if (e < N_EDGES) nrm[e] = dinv[src[e]] * dinv[dst[e]];
}

// ---------------------------------------------------------------------------
// Fp32 WMMA GEMM:  Y[N x 128] = (relu?)X[N x 128] @ W[128 x 128]
// Block = 4 waves = one 16-row strip. Each wave owns TWO 16-col tiles
// (n0, n0+16): one A fetch feeds two WMMAs per K-step (A reuse, ILP).
// RELU is a template param -> no per-iteration select.
// A layout: lane = M (0..15), half-wave selects K pair {k,k+1} vs {k+2,k+3}
// B layout: lane = N (0..15), VGPR v + half-wave*2 selects K row
// D layout: VGPR v, lane<16 -> M=v, lane>=16 -> M=v+8
// ---------------------------------------------------------------------------
template <int RELU>
__global__ __launch_bounds__(128) void gcn_gemm128_wmma(const float* __restrict__ X,
                                                        const float* __restrict__ W,
                                                        float* __restrict__ Y) {
    const int lid  = threadIdx.x & 31;
    const int wv   = threadIdx.x >> 5;      // 0..3
    const int m0   = blockIdx.x << 4;       // row tile base
    const int n0   = wv << 5;               // 2 col tiles: n0, n0+16
    const int half = lid >> 4;              // 0 or 1
    const int l15  = lid & 15;

    v8f c0 = {};
    v8f c1 = {};
    const float* arow = X + (size_t)(m0 + l15) * HDIM;
    const float* wcol = W + n0 + l15;

    for (int k = 0; k < HDIM; k += 4) {
        const int ka = k + half * 2;
        v2f a = *(const v2f*)(arow + ka);
        if (RELU) { a.x = fmaxf(a.x, 0.0f); a.y = fmaxf(a.y, 0.0f); }

        const float* wp = wcol + (size_t)ka * HDIM;
        v2f b0, b1;
        b0.x = wp[0];
        b0.y = wp[HDIM];
        b1.x = wp[16];
        b1.y = wp[HDIM + 16];

        c0 = __builtin_amdgcn_wmma_f32_16x16x4_f32(
                false, a, false, b0, (short)0, c0, false, false);
        c1 = __builtin_amdgcn_wmma_f32_16x16x4_f32(
                false, a, false, b1, (short)0, c1, false, false);
    }

    float* yrow = Y + (size_t)(m0 + half * 8) * HDIM + n0 + l15;
#pragma unroll
    for (int v = 0; v < 8; ++v) {
        yrow[(size_t)v * HDIM]      = c0[v];
        yrow[(size_t)v * HDIM + 16] = c1[v];
    }
}

// ---------------------------------------------------------------------------
// Aggregation: out[i] = b + dinv[i]^2 * hW[i]  (self-loop + bias seed)
// then edge atomics: out[dst] += norm_e * hW[src]
// ---------------------------------------------------------------------------
__global__ __launch_bounds__(256) void gcn_agg_init(const float* __restrict__ hW,
                                                    const float* __restrict__ dinv,
                                                    const float* __restrict__ bias,
                                                    float* __restrict__ out) {
    int t = blockIdx.x * 256 + threadIdx.x;     // N_NODES * HDIM threads
    if (t < N_NODES * HDIM) {
        int i = t >> 7;
        int f = t & 127;
        float dv = dinv[i];
        out[t] = bias[f] + dv * dv * hW[t];
    }
}

__global__ __launch_bounds__(256) void gcn_agg_edges(const int* __restrict__ src,
                                                     const int* __restrict__ dst,
                                                     const float* __restrict__ nrm,
                                                     const float* __restrict__ hW,
                                                     float* __restrict__ out) {
    int t    = blockIdx.x * 256 + threadIdx.x;  // N_EDGES * 32 threads
    int e    = t >> 5;
    int lane = t & 31;
    if (e < N_EDGES) {
        int   s  = src[e];
        int   d  = dst[e];
        float nr = nrm[e];
        float4 v = ((const float4*)(hW + (size_t)s * HDIM))[lane];
        float* op = out + (size_t)d * HDIM + lane * 4;
        atomicAdd(op + 0, nr * v.x);
        atomicAdd(op + 1, nr * v.y);
        atomicAdd(op + 2, nr * v.z);
        atomicAdd(op + 3, nr * v.w);
    }
}

// ---------------------------------------------------------------------------
// Mean pool over graph ids + MLP head
// ---------------------------------------------------------------------------
__global__ __launch_bounds__(256) void gcn_zero(float* __restrict__ p, int n) {
    int i = blockIdx.x * 256 + threadIdx.x;
    if (i < n) p[i] = 0.0f;
}

__global__ __launch_bounds__(256) void gcn_pool(const float* __restrict__ h,
                                                const int* __restrict__ batch,
                                                float* __restrict__ pooled,
                                                float* __restrict__ cnt) {
    int t    = blockIdx.x * 256 + threadIdx.x;  // N_NODES * 32 threads
    int i    = t >> 5;
    int lane = t & 31;
    if (i < N_NODES) {
        int b = batch[i];
        float4 v = ((const float4*)(h + (size_t)i * HDIM))[lane];
        float* pp = pooled + (size_t)b * HDIM + lane * 4;
        atomicAdd(pp + 0, v.x);
        atomicAdd(pp + 1, v.y);
        atomicAdd(pp + 2, v.z);
        atomicAdd(pp + 3, v.w);
        if (lane == 0) atomicAdd(&cnt[b], 1.0f);
    }
}

__global__ __launch_bounds__(256) void gcn_mean(const float* __restrict__ pooled,
                                                const float* __restrict__ cnt,
                                                float* __restrict__ g) {
    int t = blockIdx.x * 256 + threadIdx.x;     // NB * HDIM
    if (t < NB * HDIM) {
        int b = t >> 7;
        g[t] = pooled[t] / fmaxf(cnt[b], 1.0f);
    }
}

__global__ __launch_bounds__(64) void gcn_mlp1(const float* __restrict__ g,
                                               const float* __restrict__ fw1,
                                               const float* __restrict__ fb1,
                                               float* __restrict__ hid) {
    int b = blockIdx.x;      // 0..63
    int j = threadIdx.x;     // 0..63
    float acc = fb1[j];
    for (int k = 0; k < HDIM; ++k)
        acc += g[b * HDIM + k] * fw1[k * 64 + j];
    hid[b * 64 + j] = fmaxf(acc, 0.0f);
}

__global__ __launch_bounds__(64) void gcn_mlp2(const float* __restrict__ hid,
                                               const float* __restrict__ fw2,
                                               const float* __restrict__ fb2,
                                               float* __restrict__ out) {
    int b = threadIdx.x;     // 0..63
    float acc = fb2[0];
    for (int j = 0; j < 64; ++j)
        acc += hid[b * 64 + j] * fw2[j];
    out[b] = acc;
}

// ---------------------------------------------------------------------------
// Launch
// ---------------------------------------------------------------------------
extern "C" void kernel_launch(void* const* d_in, const int* in_sizes, int n_in,
                              void* d_out, int out_size, void* d_ws, size_t ws_size,
                              hipStream_t stream) {
    (void)in_sizes; (void)n_in; (void)out_size; (void)ws_size;

    const float* x     = (const float*)d_in[0];
    const int*   eidx  = (const int*)  d_in[1];
    const int*   batch = (const int*)  d_in[2];
    const float* W1    = (const float*)d_in[3];
    const float* b1    = (const float*)d_in[4];
    const float* W2    = (const float*)d_in[5];
    const float* b2    = (const float*)d_in[6];
    const float* W3    = (const float*)d_in[7];
    const float* b3    = (const float*)d_in[8];
    const float* fw1   = (const float*)d_in[9];
    const float* fb1   = (const float*)d_in[10];
    const float* fw2   = (const float*)d_in[11];
    const float* fb2   = (const float*)d_in[12];
    float*       out   = (float*)d_out;

    const int* src = eidx;
    const int* dst = eidx + N_EDGES;

    // Workspace carve-up (floats)
    float* ws     = (float*)d_ws;
    float* deg    = ws;                               // N
    float* dinv   = deg    + N_NODES;                 // N
    float* nrm    = dinv   + N_NODES;                 // E
    float* hW     = nrm    + N_EDGES;                 // N*H
    float* agg0   = hW     + (size_t)N_NODES * HDIM;  // N*H
    float* agg1   = agg0   + (size_t)N_NODES * HDIM;  // N*H
    float* pooled = agg1   + (size_t)N_NODES * HDIM;  // B*H
    float* cnt    = pooled + NB * HDIM;               // B  (contiguous with pooled)
    float* gbuf   = cnt    + NB;                      // B*H
    float* hid    = gbuf   + NB * HDIM;               // B*64

    const int gN   = (N_NODES + 255) / 256;           // 196
    const int gE   = (N_EDGES + 255) / 256;           // 3125
    const int gNH  = (N_NODES * HDIM) / 256;          // 25000
    const int gE32 = (N_EDGES * 32) / 256;            // 100000
    const int gN32 = (N_NODES * 32) / 256;            // 6250
    const int gRow = N_NODES / 16;                    // 3125 (exact)

    // Degrees + symmetric GCN norm
    gcn_init_deg <<<gN, 256, 0, stream>>>(deg);
    gcn_count_deg<<<gE, 256, 0, stream>>>(dst, deg);
    gcn_dinv     <<<gN, 256, 0, stream>>>(deg, dinv);
    gcn_edge_norm<<<gE, 256, 0, stream>>>(src, dst, dinv, nrm);

    // Layer 1: h = relu(A_hat (X W1) + b1)   (relu fused into next GEMM load)
    gcn_gemm128_wmma<0><<<gRow, 128, 0, stream>>>(x, W1, hW);
    gcn_agg_init       <<<gNH, 256, 0, stream>>>(hW, dinv, b1, agg0);
    gcn_agg_edges      <<<gE32, 256, 0, stream>>>(src, dst, nrm, hW, agg0);

    // Layer 2
    gcn_gemm128_wmma<1><<<gRow, 128, 0, stream>>>(agg0, W2, hW);
    gcn_agg_init       <<<gNH, 256, 0, stream>>>(hW, dinv, b2, agg1);
    gcn_agg_edges      <<<gE32, 256, 0, stream>>>(src, dst, nrm, hW, agg1);

    // Layer 3 (no relu on output)
    gcn_gemm128_wmma<1><<<gRow, 128, 0, stream>>>(agg1, W3, hW);
    gcn_agg_init       <<<gNH, 256, 0, stream>>>(hW, dinv, b3, agg0);
    gcn_agg_edges      <<<gE32, 256, 0, stream>>>(src, dst, nrm, hW, agg0);

    // Global mean pool
    gcn_zero<<<(NB * HDIM + NB + 255) / 256, 256, 0, stream>>>(pooled, NB * HDIM + NB);
    gcn_pool<<<gN32, 256, 0, stream>>>(agg0, batch, pooled, cnt);
    gcn_mean<<<(NB * HDIM + 255) / 256, 256, 0, stream>>>(pooled, cnt, gbuf);

    // MLP head
    gcn_mlp1<<<NB, 64, 0, stream>>>(gbuf, fw1, fb1, hid);
    gcn_mlp2<<<1, 64, 0, stream>>>(hid, fw2, fb2, out);
}